// SOMQuantizer_40518721470540
// MI455X (gfx1250) — compile-verified
//
#include <hip/hip_runtime.h>
#include <stdint.h>

// ---------------- problem constants ----------------
#define BS     65536
#define ZD     256
#define NN     1024
#define SOM1D  32
#define MTILE  128            // rows per workgroup (8 waves x 16 rows)
#define KB     64             // K-steps of 4 (ZD/4)
#define CT     64             // column tiles of 16 (NN/16)
#define NBUF   3              // async B-tile buffers (prefetch depth 2)
#define BDEPTH 8              // ds->wmma software pipeline depth
#define BSTRIDE 260           // padded floats per LDS B row: 1040B = 65*16B (16B aligned, bank-spread)

typedef float v2f __attribute__((ext_vector_type(2)));
typedef float v8f __attribute__((ext_vector_type(8)));

// generic -> LDS byte offset (for async-to-LDS VDST operand)
typedef __attribute__((address_space(3))) void lds_t;
static __device__ __forceinline__ uint32_t lds_off(void* p) {
    return (uint32_t)(uintptr_t)(lds_t*)p;
}

// ---------------- kernel 1: codebook row norms -> workspace ----------------
__global__ __launch_bounds__(256) void cb_norms_kernel(const float* __restrict__ cb,
                                                       float* __restrict__ ynorm) {
    const int wave = threadIdx.x >> 5;
    const int lane = threadIdx.x & 31;
    const int row  = blockIdx.x * 8 + wave;        // 128 blocks x 8 rows
    const float* p = cb + (size_t)row * ZD + lane * 8;
    float4 a = *(const float4*)p;
    float4 b = *(const float4*)(p + 4);
    float s = a.x*a.x + a.y*a.y + a.z*a.z + a.w*a.w
            + b.x*b.x + b.y*b.y + b.z*b.z + b.w*b.w;
#pragma unroll
    for (int m = 16; m >= 1; m >>= 1) s += __shfl_xor(s, m);
    if (lane == 0) ynorm[row] = s;
}

// ---------------- kernel 2: z_e passthrough copy (output 0) ----------------
__global__ __launch_bounds__(256) void copy_ze_kernel(const float* __restrict__ ze,
                                                      float* __restrict__ o) {
    size_t i = (size_t)blockIdx.x * 256 + threadIdx.x;   // in float4 units
    ((float4*)o)[i] = ((const float4*)ze)[i];
}

// ---------------- kernel 3: main SOM quantizer ----------------
__global__ __launch_bounds__(256, 1)
void som_main_kernel(const float* __restrict__ ze,
                     const float* __restrict__ cb,
                     const int*   __restrict__ epoch_p,
                     const float* __restrict__ ynorm,
                     float* __restrict__ out_k,
                     float* __restrict__ out_zq,
                     float* __restrict__ out_nw,
                     float* __restrict__ out_dist) {
    __shared__ float sB[NBUF][16 * BSTRIDE]; // triple-buffered B tile (16 codebook rows, padded)
    __shared__ float sYn[NN];                // codebook norms
    __shared__ int   sK[MTILE];              // per-row argmin indices

    const int t     = threadIdx.x;
    const int wave  = t >> 5;
    const int lane  = t & 31;
    const int half  = lane >> 4;            // K-half for A/B layout
    const int ln16  = lane & 15;            // M (for A) / N (for B,C,D) lane index
    const int rowbase = blockIdx.x * MTILE + wave * 16;
    const int myrow   = rowbase + ln16;     // A-matrix row held by this lane

    // stage y-norms into LDS
    for (int i = t; i < NN; i += 256) sYn[i] = ynorm[i];

    // ---- A tile register-resident in exact WMMA A layout, pre-scaled by -2 ----
    // lane holds K = 4*kb + 2*half + {0,1} of row `myrow`
    v2f a[KB];
    {
        const float* zp = ze + (size_t)myrow * ZD + 2 * half;
#pragma unroll
        for (int kb = 0; kb < KB; ++kb) {
            float2 v = *(const float2*)(zp + 4 * kb);
            a[kb].x = -2.0f * v.x;
            a[kb].y = -2.0f * v.y;
        }
    }
    // ||x||^2 from the resident registers: sum(a^2) = 4 * partial(x^2)
    float xs = 0.0f;
#pragma unroll
    for (int kb = 0; kb < KB; ++kb) xs += a[kb].x * a[kb].x + a[kb].y * a[kb].y;
    xs += __shfl_xor(xs, 16);
    const float xnorm_row = xs * 0.25f;     // ||x_myrow||^2
    // broadcast to C-layout: VGPR i holds row M = i + 8*half
    float xn[8];
#pragma unroll
    for (int i = 0; i < 8; ++i) xn[i] = __shfl(xnorm_row, i + 8 * half);

    // ---- async B-tile loader: codebook[ct*16 .. +16) rows, contiguous 16KB -> padded LDS ----
    const uint64_t cb64 = (uint64_t)cb;
    auto issue_b_loads = [&](int ct, int buf) {
        const uint32_t goff0   = (uint32_t)ct * 16384u;            // bytes into codebook
        const uint32_t ldsbase = lds_off(&sB[buf][0]);
#pragma unroll
        for (int it = 0; it < 4; ++it) {                           // 256 thr * 4 * 16B = 16KB
            uint32_t e16   = (uint32_t)(it * 256 + t);             // 16B-chunk index
            uint32_t row   = e16 >> 6;                             // 64 chunks per 1KB row
            uint32_t chunk = e16 & 63u;
            uint32_t lds   = ldsbase + row * (BSTRIDE * 4u) + chunk * 16u;
            uint32_t goff  = goff0 + e16 * 16u;
            asm volatile("global_load_async_to_lds_b128 %0, %1, %2"
                         :: "v"(lds), "v"(goff), "s"(cb64) : "memory");
        }
    };

    // per-lane argmin state: VGPR i tracks row M = i + 8*half, column n = ln16 + 16*ct
    float bestv[8];
    int   besti[8];
#pragma unroll
    for (int i = 0; i < 8; ++i) { bestv[i] = 3.4e38f; besti[i] = 0; }

    // prefetch two tiles ahead
    issue_b_loads(0, 0);
    issue_b_loads(1, 1);

    int buf = 0;
    for (int ct = 0; ct < CT; ++ct) {
        if (ct + 2 < CT) {
            issue_b_loads(ct + 2, (buf + 2) % NBUF);
            asm volatile("s_wait_asynccnt 8" ::: "memory");  // batch `ct` complete
        } else if (ct + 1 < CT) {
            asm volatile("s_wait_asynccnt 4" ::: "memory");
        } else {
            asm volatile("s_wait_asynccnt 0" ::: "memory");
        }
        __syncthreads();

        // B layout: lane = N (ln16), holds K = 4*kb + 2*half + {0,1} of codebook row n
        const float* bp = &sB[buf][ln16 * BSTRIDE + 2 * half];

        // depth-8 software pipeline: 8 ds_load_b64 in flight, wmma waits dscnt<=7
        v2f bb[BDEPTH];
#pragma unroll
        for (int p = 0; p < BDEPTH; ++p) {
            float2 bv = *(const float2*)(bp + 4 * p);
            bb[p].x = bv.x; bb[p].y = bv.y;
        }
        v8f c = {0.f, 0.f, 0.f, 0.f, 0.f, 0.f, 0.f, 0.f};
#pragma unroll
        for (int kb = 0; kb < KB; ++kb) {
            c = __builtin_amdgcn_wmma_f32_16x16x4_f32(
                    false, a[kb], false, bb[kb & (BDEPTH - 1)], (short)0, c, false, false);
            if (kb + BDEPTH < KB) {
                float2 bv = *(const float2*)(bp + 4 * (kb + BDEPTH));
                bb[kb & (BDEPTH - 1)].x = bv.x;
                bb[kb & (BDEPTH - 1)].y = bv.y;
            }
            __builtin_amdgcn_sched_barrier(0); // lock the pipelined load->wmma pattern
        }

        // epilogue: dist = (-2 x.y) + ||x||^2 + ||y||^2, clamp, store, track argmin
        const int n  = ct * 16 + ln16;
        const float yn = sYn[n];
#pragma unroll
        for (int i = 0; i < 8; ++i) {
            const float d = fmaxf(c[i] + xn[i] + yn, 0.0f);
            out_dist[(size_t)(rowbase + i + 8 * half) * NN + n] = d;
            if (d < bestv[i]) { bestv[i] = d; besti[i] = n; }   // strict < => first-index ties
        }
        __syncthreads();   // buffer reuse fence (protects buf (ct+2)%NBUF issued next iter)

        buf = (buf + 1) % NBUF;
    }

    // ---- cross-lane argmin within each 16-lane N-group (first-index tie-break) ----
#pragma unroll
    for (int i = 0; i < 8; ++i) {
        float v = bestv[i]; int ix = besti[i];
#pragma unroll
        for (int m = 8; m >= 1; m >>= 1) {
            float ov = __shfl_xor(v, m);
            int   oi = __shfl_xor(ix, m);
            if (ov < v || (ov == v && oi < ix)) { v = ov; ix = oi; }
        }
        bestv[i] = v; besti[i] = ix;
    }
    if (ln16 == 0) {
#pragma unroll
        for (int i = 0; i < 8; ++i) {
            const int mloc = wave * 16 + i + 8 * half;
            sK[mloc] = besti[i];
            out_k[blockIdx.x * MTILE + mloc] = (float)besti[i];
        }
    }
    __syncthreads();

    // ---- cooperative z_q gather: 128 rows x 64 float4 ----
    for (int j = t; j < MTILE * 64; j += 256) {
        const int r  = j >> 6;
        const int cq = j & 63;
        const int kk = sK[r];
        float4 v = *(const float4*)(cb + (size_t)kk * ZD + cq * 4);
        *(float4*)(out_zq + (size_t)(blockIdx.x * MTILE + r) * ZD + cq * 4) = v;
    }

    // ---- neighbour weighing: w[b,g] = exp(-((g/32 - k%32)^2 + (g%32 - k/32)^2)/(2s^2)) ----
    const int   ep     = *epoch_p;
    const float sigma  = 16.0f * exp2f(-0.05f * (float)ep);   // 16 * 32^(-e/100)
    const float inv2s2 = 1.0f / (2.0f * sigma * sigma);
    const size_t nwbase = (size_t)blockIdx.x * MTILE * NN;
#pragma unroll 4
    for (int i = 0; i < MTILE * NN / 256; ++i) {              // 512 iters, coalesced
        const int linear = i * 256 + t;
        const int r  = linear >> 10;
        const int g  = linear & 1023;
        const int kk = sK[r];
        const float dx = (float)(g >> 5) - (float)(kk & 31);  // xx - k_y
        const float dy = (float)(g & 31) - (float)(kk >> 5);  // yy - k_x
        out_nw[nwbase + linear] = expf(-(dx * dx + dy * dy) * inv2s2);
    }
}

// ---------------- launcher ----------------
extern "C" void kernel_launch(void* const* d_in, const int* in_sizes, int n_in,
                              void* d_out, int out_size, void* d_ws, size_t ws_size,
                              hipStream_t stream) {
    const float* ze = (const float*)d_in[0];
    const float* cb = (const float*)d_in[1];
    const int*   ep = (const int*)d_in[2];

    float* out    = (float*)d_out;
    float* o_ze   = out;                                   // [65536,256]
    float* o_k    = o_ze  + (size_t)BS * ZD;               // [65536]
    float* o_zq   = o_k   + BS;                            // [65536,256]
    float* o_nw   = o_zq  + (size_t)BS * ZD;               // [65536,32,32]
    float* o_dist = o_nw  + (size_t)BS * NN;               // [65536,1024]

    float* ynorm = (float*)d_ws;                           // 1024 floats scratch

    cb_norms_kernel<<<NN / 8, 256, 0, stream>>>(cb, ynorm);
    copy_ze_kernel<<<(BS * ZD / 4) / 256, 256, 0, stream>>>(ze, o_ze);
    som_main_kernel<<<BS / MTILE, 256, 0, stream>>>(ze, cb, ep, ynorm,
                                                    o_k, o_zq, o_nw, o_dist);
}